// RNN_73710228734683
// MI455X (gfx1250) — compile-verified
//
#include <hip/hip_runtime.h>
#include <hip/hip_bf16.h>

#define B_ 1024
#define E_ 20000
#define P_ 5
#define A_ 8
#define N_ 10000
#define D_ 8

#ifndef __has_builtin
#define __has_builtin(x) 0
#endif

#if __has_builtin(__builtin_amdgcn_global_load_async_to_lds_b128)
#define USE_ASYNC_LDS 1
#else
#define USE_ASYNC_LDS 0
#endif

// The builtin's pointer parameters are 16-byte int vectors (per hipcc diag).
typedef int v4i __attribute__((vector_size(16)));

__device__ __forceinline__ void wait_async_zero() {
#if __has_builtin(__builtin_amdgcn_s_wait_asynccnt)
    __builtin_amdgcn_s_wait_asynccnt(0);
#else
    asm volatile("s_wait_asynccnt 0" ::: "memory");
#endif
}

// ---------------------------------------------------------------------------
// Kernel 1: per-(b,dim) row constant C = max(x) + log(sum(exp(x - max))).
// One block per row (5120 rows of E_=20000 contiguous fp32).
// Row is staged global->LDS via CDNA5 async-to-LDS (ASYNCcnt), then reduced
// twice (max, sum-exp) from LDS -> exactly one HBM pass per row.
// LDS: 80 KB/row -> 4 blocks resident per WGP (320 KB), 32 waves/WGP.
// ---------------------------------------------------------------------------
__global__ void RNN_rowlse_kernel(const float* __restrict__ pred,
                                  float* __restrict__ rowC) {
    extern __shared__ float smem[];      // E_ floats + 16 reduction slots
    float* red = smem + E_;

    const float* row = pred + (size_t)blockIdx.x * (size_t)E_;
    const int t    = threadIdx.x;
    const int nthr = blockDim.x;

#if USE_ASYNC_LDS
    // Per-lane async DMA: 16B global -> 16B LDS, tracked by ASYNCcnt.
    for (int i = t; i < E_ / 4; i += nthr) {
        __builtin_amdgcn_global_load_async_to_lds_b128(
            (v4i*)(row + 4 * i), (v4i*)(smem + 4 * i), 0, 0);
    }
    wait_async_zero();                   // this wave's LDS writes landed
#else
    for (int i = t; i < E_ / 4; i += nthr) {
        const float4 v = ((const float4*)row)[i];
        ((float4*)smem)[i] = v;          // global_load_b128 + ds_store_b128
    }
#endif
    __syncthreads();

    // ---- pass 1: row max ----
    float m = -__builtin_inff();
    for (int i = t; i < E_; i += nthr) m = fmaxf(m, smem[i]);
    for (int off = 16; off > 0; off >>= 1)
        m = fmaxf(m, __shfl_xor(m, off, 32));
    const int wave = t >> 5;
    const int lane = t & 31;
    if (lane == 0) red[wave] = m;
    __syncthreads();
    if (t == 0) {
        float mm = red[0];
        for (int w = 1; w < (nthr >> 5); ++w) mm = fmaxf(mm, red[w]);
        red[8] = mm;
    }
    __syncthreads();
    const float rowmax = red[8];

    // ---- pass 2: sum of exp(x - max) ----
    float s = 0.0f;
    for (int i = t; i < E_; i += nthr) s += __expf(smem[i] - rowmax);
    for (int off = 16; off > 0; off >>= 1)
        s += __shfl_xor(s, off, 32);
    if (lane == 0) red[wave] = s;
    __syncthreads();
    if (t == 0) {
        float ss = red[0];
        for (int w = 1; w < (nthr >> 5); ++w) ss += red[w];
        rowC[blockIdx.x] = rowmax + __logf(ss);
    }
}

// ---------------------------------------------------------------------------
// Kernel 2: greedy sequential chain, one thread per batch element.
// Gathers 8 adjacent-edge logits per step; logp[e] = pred[e] - C(b,dim).
// ---------------------------------------------------------------------------
__global__ void RNN_chain_kernel(const float* __restrict__ pred,
                                 const float* __restrict__ pred_d,
                                 const int*   __restrict__ gt,
                                 const int*   __restrict__ obs,
                                 const int*   __restrict__ graph_edges,
                                 const int*   __restrict__ node_adj_edges,
                                 const int*   __restrict__ offset_p,
                                 const float* __restrict__ rowC,
                                 float*       __restrict__ out) {
    const int b = blockIdx.x * blockDim.x + threadIdx.x;
    if (b >= B_) return;

    const int offset = offset_p[0];
    int end_node  = graph_edges[gt[b * P_ + 0] * 3 + 0] - 1;   // node ids 1..N
    int last_pred = obs[b];

    float* out_pred   = out;                 // (B,P) as float
    float* out_pred_d = out + B_ * P_;       // (B,P) as float
    float* out_scores = out + 2 * B_ * P_;   // (B,P,2)

    for (int dim = 0; dim < P_; ++dim) {
        const float  C    = rowC[b * P_ + dim];
        const float* prow = pred + ((size_t)b * P_ + (size_t)dim) * (size_t)E_;

        // adjacency gather, mask the edge we came from, padded (==E_) -> -inf
        int   nae[A_];
        float best = -__builtin_inff();
        int   bidx = 0;
#pragma unroll
        for (int j = 0; j < A_; ++j) {
            int e = node_adj_edges[end_node * A_ + j];
            if (e == last_pred) e = E_;
            nae[j] = e;
            const float v = (e == E_) ? -__builtin_inff() : (prow[e] - C);
            if (v > best) { best = v; bidx = j; }   // first-max, like argmax
        }
        int cur = nae[bidx];
        if (cur == E_) cur = E_ - offset;           // padding fallback
        last_pred = cur;
        end_node  = graph_edges[cur * 3 + 1] - 1;

        // direction head: argmax + its log-softmax value = -log(sum(exp(x-max)))
        const float* drow = pred_d + (size_t)b * (D_ * P_) + (size_t)dim * D_;
        float dmax = drow[0];
        int   didx = 0;
#pragma unroll
        for (int j = 1; j < D_; ++j) {
            const float v = drow[j];
            if (v > dmax) { dmax = v; didx = j; }
        }
        float dsum = 0.0f;
#pragma unroll
        for (int j = 0; j < D_; ++j) dsum += __expf(drow[j] - dmax);
        const float d_best = -__logf(dsum);

        out_pred  [b * P_ + dim] = (float)cur;
        out_pred_d[b * P_ + dim] = (float)didx;
        out_scores[(b * P_ + dim) * 2 + 0] = best;
        out_scores[(b * P_ + dim) * 2 + 1] = d_best;
    }
}

extern "C" void kernel_launch(void* const* d_in, const int* in_sizes, int n_in,
                              void* d_out, int out_size, void* d_ws, size_t ws_size,
                              hipStream_t stream) {
    const float* pred        = (const float*)d_in[0];
    const float* pred_d      = (const float*)d_in[1];
    const int*   gt          = (const int*)  d_in[2];
    const int*   obs         = (const int*)  d_in[3];
    const int*   graph_edges = (const int*)  d_in[4];
    const int*   node_adj    = (const int*)  d_in[5];
    const int*   offset      = (const int*)  d_in[6];
    float*       out         = (float*)d_out;
    float*       rowC        = (float*)d_ws;          // 5120 floats scratch

    const size_t smem_bytes = (size_t)(E_ + 16) * sizeof(float);   // ~80 KB

    RNN_rowlse_kernel<<<B_ * P_, 256, smem_bytes, stream>>>(pred, rowC);
    RNN_chain_kernel<<<(B_ + 255) / 256, 256, 0, stream>>>(
        pred, pred_d, gt, obs, graph_edges, node_adj, offset, rowC, out);
}